// GATEncoder_84473416777940
// MI455X (gfx1250) — compile-verified
//
#include <hip/hip_runtime.h>
#include <hip/hip_bf16.h>

typedef __attribute__((ext_vector_type(2))) float v2f;
typedef __attribute__((ext_vector_type(8))) float v8f;

#define NEG_SLOPE 0.2f

// ---------- helpers: monotone float<->int encoding so signed atomicMax == float max ----------
__device__ __forceinline__ int enc_f32(float f) {
  int i = __float_as_int(f);
  return i < 0 ? (int)(0x80000000u - (unsigned)i) : i;
}
__device__ __forceinline__ float dec_f32(int e) {
  int i = e < 0 ? (int)(0x80000000u - (unsigned)e) : e;
  return __int_as_float(i);
}

// ---------- generic 32-bit fill (zeroing + (-inf) init; graph-capture-safe) ----------
__global__ void fill_u32(unsigned int* __restrict__ p, int n, unsigned int v) {
  int i = blockIdx.x * blockDim.x + threadIdx.x;
  if (i < n) p[i] = v;
}

// ---------- fp32 WMMA GEMM: C[M,N] = A[M,K] @ B[K,N], row-major ----------
// Block: 256 threads (8 waves) covers 8 M-tiles x one 64-column slab.
// B slab staged in LDS, K-pair interleaved + parity-XOR bank swizzle:
//   element (row, col) -> sB[pair*128 + (col ^ ((pair&1)<<4))*2 + (row&1)], pair = row/2.
// Each WMMA B operand {B[k][c], B[k+1][c]} is then one aligned ds_load_b64, and the two
// wave-halves (reading pairs p, p+1 of opposite parity) land on disjoint 32-bank halves
// -> conflict-free 64-bank access, no repacking v_mov's.
// Each wave: 16-row strip, 4 accumulators (16x64 output), K-loop of v_wmma_f32_16x16x4_f32.
__global__ __launch_bounds__(256) void gemm_wmma_f32(
    const float* __restrict__ A, const float* __restrict__ B, float* __restrict__ C,
    int M, int N, int K) {
  __shared__ float sB[128 * 128];  // (K/2 <= 128 pairs) x 128 floats = 64 KB
  const int tid = threadIdx.x;
  const int c0 = blockIdx.y * 64;

  // cooperative stage: float4 global reads, swizzled scalar LDS stores
  const int nvec = K * 16;  // float4s in slab
  for (int i = tid; i < nvec; i += 256) {
    int row = i >> 4, q = i & 15;
    float4 v = *(const float4*)(B + (size_t)row * N + c0 + q * 4);
    int pair = row >> 1;
    int x = (pair & 1) << 4;
    float* dstp = sB + pair * 128 + (((q * 4) ^ x) << 1) + (row & 1);
    dstp[0] = v.x;
    dstp[2] = v.y;
    dstp[4] = v.z;
    dstp[6] = v.w;
  }
  __syncthreads();

  const int wave = tid >> 5, lane = tid & 31;
  const int tile_m = blockIdx.x * 8 + wave;
  if (tile_m * 16 >= M) return;  // wave-uniform: EXEC all-ones inside WMMA
  const int hi = lane >> 4;      // half of the wave selects K pair
  const int l16 = lane & 15;

  const float* Abase = A + (size_t)(tile_m * 16 + l16) * K + hi * 2;
  v8f acc0 = {}, acc1 = {}, acc2 = {}, acc3 = {};
  for (int k = 0; k < K; k += 4) {
    v2f a;
    a.x = Abase[k];      // contiguous pair -> single global_load_b64
    a.y = Abase[k + 1];
    const int pair = (k >> 1) + hi;        // K-row pair this half consumes
    const int x = (pair & 1) << 4;         // bank swizzle
    const float* sp = sB + pair * 128;
    v2f b0 = *(const v2f*)(sp + ((l16 + 0) ^ x) * 2);    // cols l16
    v2f b1 = *(const v2f*)(sp + ((l16 + 16) ^ x) * 2);   // cols l16+16
    v2f b2 = *(const v2f*)(sp + ((l16 + 32) ^ x) * 2);   // cols l16+32
    v2f b3 = *(const v2f*)(sp + ((l16 + 48) ^ x) * 2);   // cols l16+48
    acc0 = __builtin_amdgcn_wmma_f32_16x16x4_f32(false, a, false, b0, (short)0, acc0, false, false);
    acc1 = __builtin_amdgcn_wmma_f32_16x16x4_f32(false, a, false, b1, (short)0, acc1, false, false);
    acc2 = __builtin_amdgcn_wmma_f32_16x16x4_f32(false, a, false, b2, (short)0, acc2, false, false);
    acc3 = __builtin_amdgcn_wmma_f32_16x16x4_f32(false, a, false, b3, (short)0, acc3, false, false);
  }
  float* Cp = C + (size_t)(tile_m * 16 + hi * 8) * N + c0 + l16;
#pragma unroll
  for (int r = 0; r < 8; ++r) {
    Cp[(size_t)r * N + 0]  = acc0[r];
    Cp[(size_t)r * N + 16] = acc1[r];
    Cp[(size_t)r * N + 32] = acc2[r];
    Cp[(size_t)r * N + 48] = acc3[r];
  }
}

// ---------- per-(node,head) attention scalars ----------
__global__ void attn_scores(const float* __restrict__ h, const float* __restrict__ a_src,
                            const float* __restrict__ a_dst, float* __restrict__ as,
                            float* __restrict__ ad, int n, int H, int D) {
  int idx = blockIdx.x * blockDim.x + threadIdx.x;
  if (idx >= n * H) return;
  int node = idx / H, hd = idx - node * H;
  const float* hp = h + (size_t)node * H * D + hd * D;
  const float* sp = a_src + hd * D;
  const float* dp = a_dst + hd * D;
  float s = 0.f, d = 0.f;
  for (int i = 0; i < D; ++i) {
    float v = hp[i];
    s += v * sp[i];
    d += v * dp[i];
  }
  as[idx] = s;
  ad[idx] = d;
}

// ---------- segment max of leaky-relu logits over incoming edges of dst ----------
__global__ void edge_logit_max(const int* __restrict__ srcE, const int* __restrict__ dstE,
                               int E, int n, const float* __restrict__ as,
                               const float* __restrict__ ad, int* __restrict__ mEnc, int H) {
  int idx = blockIdx.x * blockDim.x + threadIdx.x;
  int tot = (E + n) * H;
  if (idx >= tot) return;
  int e = idx / H, hd = idx - e * H;
  int s, d;
  if (e < E) { s = srcE[e]; d = dstE[e]; } else { s = d = e - E; }  // self loops
  float l = as[s * H + hd] + ad[d * H + hd];
  l = l > 0.f ? l : l * NEG_SLOPE;
  atomicMax(&mEnc[d * H + hd], enc_f32(l));
}

// ---------- e = exp(logit - m[dst]); store per-edge, accumulate denom[dst] ----------
__global__ void edge_exp_sum(const int* __restrict__ srcE, const int* __restrict__ dstE,
                             int E, int n, const float* __restrict__ as,
                             const float* __restrict__ ad, const int* __restrict__ mEnc,
                             float* __restrict__ eBuf, float* __restrict__ denom, int H) {
  int idx = blockIdx.x * blockDim.x + threadIdx.x;
  int tot = (E + n) * H;
  if (idx >= tot) return;
  int e = idx / H, hd = idx - e * H;
  int s, d;
  if (e < E) { s = srcE[e]; d = dstE[e]; } else { s = d = e - E; }
  float l = as[s * H + hd] + ad[d * H + hd];
  l = l > 0.f ? l : l * NEG_SLOPE;
  float m = dec_f32(mEnc[d * H + hd]);
  float ev = __expf(l - m);
  eBuf[idx] = ev;
  atomicAdd(&denom[d * H + hd], ev);
}

// ---------- out[dst] += h[src] * alpha  (one thread per edge-feature) ----------
__global__ void edge_aggregate(const int* __restrict__ srcE, const int* __restrict__ dstE,
                               int E, int n, const float* __restrict__ h,
                               const float* __restrict__ eBuf, const float* __restrict__ denom,
                               float* __restrict__ out, int H, int D) {
  long long idx = (long long)blockIdx.x * blockDim.x + threadIdx.x;
  long long tot = (long long)(E + n) * H * D;
  if (idx >= tot) return;
  int HD = H * D;
  int f = (int)(idx % HD);
  int e = (int)(idx / HD);
  int hd = f / D;
  int s, d;
  if (e < E) { s = srcE[e]; d = dstE[e]; } else { s = d = e - E; }
  const float* hsrc = h + (size_t)s * HD + f;
  __builtin_prefetch(hsrc, 0, 1);  // global_prefetch_b8: issue gather line early
  float alpha = eBuf[(size_t)e * H + hd] / (denom[d * H + hd] + 1e-16f);
  atomicAdd(&out[(size_t)d * HD + f], *hsrc * alpha);
}

// ---------- in-place bias + ELU ----------
__global__ void bias_elu(float* __restrict__ buf, const float* __restrict__ bias, int total,
                         int F) {
  int i = blockIdx.x * blockDim.x + threadIdx.x;
  if (i >= total) return;
  float v = buf[i] + bias[i % F];
  buf[i] = v > 0.f ? v : (__expf(v) - 1.f);
}

// ---------- mean over heads + bias -> final output ----------
__global__ void head_mean_bias(const float* __restrict__ agg, const float* __restrict__ bias,
                               float* __restrict__ out, int n, int H, int D) {
  int i = blockIdx.x * blockDim.x + threadIdx.x;
  if (i >= n * D) return;
  int node = i / D, dd = i - node * D;
  float s = 0.f;
  for (int hh = 0; hh < H; ++hh) s += agg[(size_t)node * H * D + hh * D + dd];
  out[i] = s * (1.f / H) + bias[dd];
}

extern "C" void kernel_launch(void* const* d_in, const int* in_sizes, int n_in, void* d_out,
                              int out_size, void* d_ws, size_t ws_size, hipStream_t stream) {
  const float* x = (const float*)d_in[0];
  const int* ei = (const int*)d_in[1];
  const float* W1 = (const float*)d_in[2];
  const float* a_src1 = (const float*)d_in[3];
  const float* a_dst1 = (const float*)d_in[4];
  const float* b1 = (const float*)d_in[5];
  const float* W2 = (const float*)d_in[6];
  const float* a_src2 = (const float*)d_in[7];
  const float* a_dst2 = (const float*)d_in[8];
  const float* b2 = (const float*)d_in[9];
  float* out = (float*)d_out;

  const int IN_DIM = 128, HID = 256, LAT = 64, H = 4, D1 = 64;
  const int N = in_sizes[0] / IN_DIM;   // 50000 (multiple of 16)
  const int E = in_sizes[1] / 2;        // 800000
  const int Etot = E + N;               // self loops appended
  const int* srcE = ei;
  const int* dstE = ei + E;

  // workspace carve-up (~123 MB)
  char* w = (char*)d_ws;
  float* bufA = (float*)w;  w += (size_t)N * HID * 4;   // h1, then h2
  float* bufB = (float*)w;  w += (size_t)N * HID * 4;   // agg1/out1, then agg2
  float* eBuf = (float*)w;  w += (size_t)Etot * H * 4;  // per-edge exp weights
  float* as   = (float*)w;  w += (size_t)N * H * 4;
  float* ad   = (float*)w;  w += (size_t)N * H * 4;
  int*   mEnc = (int*)w;    w += (size_t)N * H * 4;
  float* denom= (float*)w;  w += (size_t)N * H * 4;

  const unsigned ENC_NEG_INF = 0x80800000u;  // enc_f32(-inf)
  const int TB = 256;
  auto blocks = [](long long total, int tb) { return (unsigned)((total + tb - 1) / tb); };

  const int mgroups = (N / 16 + 7) / 8;            // 8 M-tiles per block
  dim3 ggrid(mgroups, HID / 64);                   // 64-col slabs

  // ===== layer 1 =====
  gemm_wmma_f32<<<ggrid, TB, 0, stream>>>(x, W1, bufA, N, HID, IN_DIM);
  attn_scores<<<blocks((long long)N * H, TB), TB, 0, stream>>>(bufA, a_src1, a_dst1, as, ad, N,
                                                               H, D1);
  fill_u32<<<blocks((long long)N * H, TB), TB, 0, stream>>>((unsigned*)mEnc, N * H, ENC_NEG_INF);
  fill_u32<<<blocks((long long)N * H, TB), TB, 0, stream>>>((unsigned*)denom, N * H, 0u);
  edge_logit_max<<<blocks((long long)Etot * H, TB), TB, 0, stream>>>(srcE, dstE, E, N, as, ad,
                                                                     mEnc, H);
  edge_exp_sum<<<blocks((long long)Etot * H, TB), TB, 0, stream>>>(srcE, dstE, E, N, as, ad,
                                                                   mEnc, eBuf, denom, H);
  fill_u32<<<blocks((long long)N * HID, TB), TB, 0, stream>>>((unsigned*)bufB, N * HID, 0u);
  edge_aggregate<<<blocks((long long)Etot * H * D1, TB), TB, 0, stream>>>(
      srcE, dstE, E, N, bufA, eBuf, denom, bufB, H, D1);
  bias_elu<<<blocks((long long)N * HID, TB), TB, 0, stream>>>(bufB, b1, N * HID, HID);

  // ===== layer 2 =====
  gemm_wmma_f32<<<ggrid, TB, 0, stream>>>(bufB, W2, bufA, N, HID, HID);
  attn_scores<<<blocks((long long)N * H, TB), TB, 0, stream>>>(bufA, a_src2, a_dst2, as, ad, N,
                                                               H, LAT);
  fill_u32<<<blocks((long long)N * H, TB), TB, 0, stream>>>((unsigned*)mEnc, N * H, ENC_NEG_INF);
  fill_u32<<<blocks((long long)N * H, TB), TB, 0, stream>>>((unsigned*)denom, N * H, 0u);
  edge_logit_max<<<blocks((long long)Etot * H, TB), TB, 0, stream>>>(srcE, dstE, E, N, as, ad,
                                                                     mEnc, H);
  edge_exp_sum<<<blocks((long long)Etot * H, TB), TB, 0, stream>>>(srcE, dstE, E, N, as, ad,
                                                                   mEnc, eBuf, denom, H);
  fill_u32<<<blocks((long long)N * HID, TB), TB, 0, stream>>>((unsigned*)bufB, N * HID, 0u);
  edge_aggregate<<<blocks((long long)Etot * H * LAT, TB), TB, 0, stream>>>(
      srcE, dstE, E, N, bufA, eBuf, denom, bufB, H, LAT);
  head_mean_bias<<<blocks((long long)N * LAT, TB), TB, 0, stream>>>(bufB, b2, out, N, H, LAT);
}